// GCNnet_67267777790472
// MI455X (gfx1250) — compile-verified
//
#include <hip/hip_runtime.h>

// CDNA5 / gfx1250 GCN forward:
//   XW1 = feature @ W1                 (WMMA f32 16x16x4, LDS-staged; NT loads on the 1.15GB stream)
//   H   = spmm(A, XW1) + b1      (edge-parallel f32 atomics, L2-resident targets)
//   HW2 = relu(H) @ W2                 (WMMA f32 16x16x4, ReLU fused on A-read)
//   out = spmm(A, HW2) + b2            (edge-parallel f32 atomics)

typedef __attribute__((ext_vector_type(2))) float v2f;
typedef __attribute__((ext_vector_type(8))) float v8f;

#define N_NODES 200000
#define N_EDGES 6400000
#define IN_DIM  1433
#define HID     16
#define OUT_DIM 7

#define KC 32               // K-chunk staged through LDS
#define ROWS_PER_BLOCK 128  // 8 wave32 tiles of 16 rows
#define LDA_PAD 34          // padded LDS row stride (even -> 8B aligned float2, bank-spread)

// ---------------------------------------------------------------- GEMM1
__global__ __launch_bounds__(256) void gemm1_kernel(const float* __restrict__ X,
                                                    const float* __restrict__ W1,
                                                    float* __restrict__ XW1) {
    __shared__ float As[ROWS_PER_BLOCK * LDA_PAD]; // 17408 B
    __shared__ float Bs[KC * HID];                 //  2048 B

    const int tid  = threadIdx.x;
    const int lane = tid & 31;
    const int wave = tid >> 5;
    const int rowBase = blockIdx.x * ROWS_PER_BLOCK;
    const int tileRow = rowBase + wave * 16;        // wave-uniform
    const bool haveTile = (tileRow < N_NODES);      // wave-uniform -> EXEC all-1s in WMMA

    v8f acc = {};

    const int nChunks = (IN_DIM + KC - 1) / KC;     // 45
    for (int ch = 0; ch < nChunks; ++ch) {
        const int k0 = ch * KC;

        // Cooperative, coalesced staging: A chunk 128x32 (+zero pad), B chunk 32x16.
        // X is stream-once (1.15 GB): non-temporal so it doesn't evict the
        // L2-resident XW1/H working set used by the SpMM phases.
        for (int i = tid; i < ROWS_PER_BLOCK * KC; i += 256) {
            int r  = i >> 5;          // 0..127
            int c  = i & (KC - 1);    // 0..31  (consecutive tid -> consecutive cols)
            int gr = rowBase + r;
            int gk = k0 + c;
            float v = 0.0f;
            if (gr < N_NODES && gk < IN_DIM)
                v = __builtin_nontemporal_load(&X[(long long)gr * IN_DIM + gk]);
            As[r * LDA_PAD + c] = v;
        }
        for (int i = tid; i < KC * HID; i += 256) {
            int kk = i >> 4;
            int n  = i & 15;
            int gk = k0 + kk;
            Bs[kk * HID + n] = (gk < IN_DIM) ? W1[gk * HID + n] : 0.0f;
        }
        __syncthreads();

        if (haveTile) {
            const int m    = lane & 15;
            const int koff = (lane >> 4) * 2;       // lanes 0-15: K+0/K+1, lanes 16-31: K+2/K+3
            const float* arow = &As[(wave * 16 + m) * LDA_PAD];
            #pragma unroll
            for (int kk = 0; kk < KC; kk += 4) {
                v2f a, b;
                a.x = arow[kk + koff];
                a.y = arow[kk + koff + 1];
                b.x = Bs[(kk + koff) * HID + m];
                b.y = Bs[(kk + koff + 1) * HID + m];
                acc = __builtin_amdgcn_wmma_f32_16x16x4_f32(
                        false, a, false, b, (short)0, acc, false, false);
            }
        }
        __syncthreads();
    }

    if (haveTile) {
        const int n     = lane & 15;
        const int mhalf = (lane >> 4) * 8;          // C/D layout: VGPR r -> M=r (lo) / M=r+8 (hi)
        #pragma unroll
        for (int r = 0; r < 8; ++r) {
            XW1[(long long)(tileRow + mhalf + r) * HID + n] = acc[r];
        }
    }
}

// ---------------------------------------------------------------- bias inits
__global__ __launch_bounds__(256) void init_h_kernel(float* __restrict__ H,
                                                     const float* __restrict__ b1) {
    int i = blockIdx.x * 256 + threadIdx.x;
    if (i < N_NODES * HID) H[i] = b1[i & 15];
}

__global__ __launch_bounds__(256) void init_out_kernel(float* __restrict__ O,
                                                       const float* __restrict__ b2) {
    int i = blockIdx.x * 256 + threadIdx.x;
    if (i < N_NODES * OUT_DIM) O[i] = b2[i % OUT_DIM];
}

// ---------------------------------------------------------------- SpMM 1 (16 lanes / edge)
__global__ __launch_bounds__(256) void spmm1_kernel(const long long* __restrict__ rows,
                                                    const long long* __restrict__ cols,
                                                    const float* __restrict__ vals,
                                                    const float* __restrict__ XW1,
                                                    float* __restrict__ H) {
    long long t = (long long)blockIdx.x * 256 + threadIdx.x;
    long long e = t >> 4;
    int j = (int)(t & 15);
    if (e >= N_EDGES) return;
    // Edge streams are read-once: non-temporal keeps XW1/H hot in L2.
    int r   = (int)__builtin_nontemporal_load(&rows[e]);
    int c   = (int)__builtin_nontemporal_load(&cols[e]);
    float v = __builtin_nontemporal_load(&vals[e]);
    atomicAdd(&H[(long long)r * HID + j], v * XW1[(long long)c * HID + j]);
}

// ---------------------------------------------------------------- GEMM2 (ReLU fused on A)
__global__ __launch_bounds__(256) void gemm2_kernel(const float* __restrict__ H,
                                                    const float* __restrict__ W2,
                                                    float* __restrict__ HW2) {
    __shared__ float Ws[16 * 16];                   // W2 zero-padded to 16x16
    const int tid = threadIdx.x;
    {
        int k = tid >> 4, n = tid & 15;
        Ws[tid] = (n < OUT_DIM) ? W2[k * OUT_DIM + n] : 0.0f;
    }
    __syncthreads();

    const int lane = tid & 31;
    const int wave = tid >> 5;
    const int tileRow = blockIdx.x * ROWS_PER_BLOCK + wave * 16;  // wave-uniform
    if (tileRow >= N_NODES) return;

    const int m    = lane & 15;
    const int koff = (lane >> 4) * 2;
    const float* hrow = &H[(long long)(tileRow + m) * HID];

    v8f acc = {};
    #pragma unroll
    for (int kk = 0; kk < HID; kk += 4) {
        v2f a, b;
        a.x = fmaxf(hrow[kk + koff], 0.0f);         // fused ReLU
        a.y = fmaxf(hrow[kk + koff + 1], 0.0f);
        b.x = Ws[(kk + koff) * 16 + m];
        b.y = Ws[(kk + koff + 1) * 16 + m];
        acc = __builtin_amdgcn_wmma_f32_16x16x4_f32(
                false, a, false, b, (short)0, acc, false, false);
    }

    const int n     = lane & 15;
    const int mhalf = (lane >> 4) * 8;
    if (n < OUT_DIM) {
        #pragma unroll
        for (int r = 0; r < 8; ++r) {
            HW2[(long long)(tileRow + mhalf + r) * OUT_DIM + n] = acc[r];
        }
    }
}

// ---------------------------------------------------------------- SpMM 2 (8 lanes / edge)
__global__ __launch_bounds__(256) void spmm2_kernel(const long long* __restrict__ rows,
                                                    const long long* __restrict__ cols,
                                                    const float* __restrict__ vals,
                                                    const float* __restrict__ HW2,
                                                    float* __restrict__ O) {
    long long t = (long long)blockIdx.x * 256 + threadIdx.x;
    long long e = t >> 3;
    int j = (int)(t & 7);
    if (e >= N_EDGES || j >= OUT_DIM) return;
    int r   = (int)__builtin_nontemporal_load(&rows[e]);
    int c   = (int)__builtin_nontemporal_load(&cols[e]);
    float v = __builtin_nontemporal_load(&vals[e]);
    atomicAdd(&O[(long long)r * OUT_DIM + j], v * HW2[(long long)c * OUT_DIM + j]);
}

// ---------------------------------------------------------------- launch
extern "C" void kernel_launch(void* const* d_in, const int* in_sizes, int n_in,
                              void* d_out, int out_size, void* d_ws, size_t ws_size,
                              hipStream_t stream) {
    const long long* adj  = (const long long*)d_in[0];   // [2, E] int64, row-major
    const long long* rows = adj;
    const long long* cols = adj + N_EDGES;
    const float* vals = (const float*)d_in[1];
    const float* X    = (const float*)d_in[2];
    const float* W1   = (const float*)d_in[3];
    const float* b1   = (const float*)d_in[4];
    const float* W2   = (const float*)d_in[5];
    const float* b2   = (const float*)d_in[6];
    float* out = (float*)d_out;

    float* XW1 = (float*)d_ws;                                 // 200000*16 f32
    float* H   = XW1 + (size_t)N_NODES * HID;                  // 200000*16 f32
    float* HW2 = H   + (size_t)N_NODES * HID;                  // 200000*7  f32

    const int nTiles     = N_NODES / 16;                       // 12500
    const int gemmBlocks = (nTiles + 7) / 8;                   // 1563

    gemm1_kernel<<<gemmBlocks, 256, 0, stream>>>(X, W1, XW1);

    init_h_kernel<<<(N_NODES * HID + 255) / 256, 256, 0, stream>>>(H, b1);
    spmm1_kernel<<<(int)(((long long)N_EDGES * 16 + 255) / 256), 256, 0, stream>>>(
        rows, cols, vals, XW1, H);

    gemm2_kernel<<<gemmBlocks, 256, 0, stream>>>(H, W2, HW2);

    init_out_kernel<<<(N_NODES * OUT_DIM + 255) / 256, 256, 0, stream>>>(out, b2);
    spmm2_kernel<<<(int)(((long long)N_EDGES * 8 + 255) / 256), 256, 0, stream>>>(
        rows, cols, vals, HW2, out);
}